// SinkhornKnopp_im_75411035783865
// MI455X (gfx1250) — compile-verified
//
#include <hip/hip_runtime.h>

// Sinkhorn-Knopp, algebraically reduced to 4 streaming passes over logits
// (1M x 128 f32, 512MB) + 1 output write: ~2.5GB -> ~110us at 23.3 TB/s HBM.
// Row-sum passes run in a log2-shifted f32 domain (values in [0,1]) so the
// 16x16 row reductions can legally use v_wmma_f32_16x16x4_f32 (B = ones).
// Tile staging uses global_load_async_to_lds_b128 (ASYNCcnt). All cross-block
// reductions are two-stage and deterministic (no float atomics).
// exp2 uses raw v_exp_f32 (__builtin_amdgcn_exp2f): flush-below-2^-126 is
// harmless here since every summand is <= 1 and sums are dominated by O(1) terms.

typedef __attribute__((ext_vector_type(2))) float v2f;
typedef __attribute__((ext_vector_type(8))) float v8f;

#define NBLK 1024
#define THREADS 256
#define B_DIM (1024 * 1024)
#define K_DIM 128
#define N_TILES (B_DIM / 16)   // 65536 tiles of 16 rows
#define N_CHUNKS (B_DIM / 8)   // 131072 chunks of 8 rows
#define N_WAVES (NBLK * 8)     // 8192 wave32s
#define PITCH 68               // LDS row pitch in floats (16B aligned, conflict-free)

// raw v_exp_f32 (base-2), no denormal guard
static __device__ __forceinline__ float fast_exp2(float x) {
  return __builtin_amdgcn_exp2f(x);
}

// exp(20*x) in full f64 range: q = x*20/ln2; 2^frac via v_exp_f32, 2^int via exponent bits.
static __device__ __forceinline__ double exp20_f64(float x) {
  const double K1 = 28.853900817779268;  // 20 / ln(2)
  double y = (double)x * K1;
  double n = rint(y);
  float r = (float)(y - n);          // |r| <= 0.5, guard never needed
  float f = fast_exp2(r);
  long long e = (long long)n + 1023;
  double sc = __longlong_as_double(e << 52);
  return (double)f * sc;
}

// --- softmax(w) in f64 (128 elements, trivial) ---
__global__ void k_softmax_w(const float* __restrict__ w, double* __restrict__ k2d) {
  if (threadIdx.x == 0) {
    double m = -1e300;
    for (int k = 0; k < K_DIM; ++k) m = fmax(m, (double)w[k]);
    double s = 0.0;
    for (int k = 0; k < K_DIM; ++k) s += exp((double)w[k] - m);
    for (int k = 0; k < K_DIM; ++k) k2d[k] = exp((double)w[k] - m) / s;
  }
}

// --- Pass 0: R0[k] = sum_b exp(20*L[b,k]) in f64 (range needs f64). ---
__global__ __launch_bounds__(THREADS) void k_colsum0(const float* __restrict__ logits,
                                                     double* __restrict__ partial) {
  __shared__ double red[8][128];
  const int w = threadIdx.x >> 5;
  const int l = threadIdx.x & 31;
  const int wgid = blockIdx.x * 8 + w;
  double a0 = 0, a1 = 0, a2 = 0, a3 = 0;
  for (int c = wgid; c < N_CHUNKS; c += N_WAVES) {
    const float* gp = logits + (size_t)c * (8 * K_DIM) + l * 4;
#pragma unroll
    for (int i = 0; i < 8; ++i) {
      float4 v = *(const float4*)(gp + i * K_DIM);
      a0 += exp20_f64(v.x);
      a1 += exp20_f64(v.y);
      a2 += exp20_f64(v.z);
      a3 += exp20_f64(v.w);
    }
  }
  red[w][4 * l + 0] = a0;
  red[w][4 * l + 1] = a1;
  red[w][4 * l + 2] = a2;
  red[w][4 * l + 3] = a3;
  __syncthreads();
  if (threadIdx.x < 128) {
    double s = 0.0;
#pragma unroll
    for (int i = 0; i < 8; ++i) s += red[i][threadIdx.x];
    partial[(size_t)blockIdx.x * 128 + threadIdx.x] = s;
  }
}

// qa0[k] = log2(k2[k]/R0[k])
__global__ void k_make_qa0(const double* __restrict__ partial, const double* __restrict__ k2d,
                           float* __restrict__ qa0) {
  int k = threadIdx.x;
  double s = 0.0;
  for (int b = 0; b < NBLK; ++b) s += partial[(size_t)b * 128 + k];
  qa0[k] = (float)(log2(k2d[k]) - log2(s));
}

// qaNext[k] = log2(k2[k]*aPrev[k]/T[k]) where T[k] = sum_b s/ctilde = a_prev[k]*R_next[k]
__global__ void k_make_qa_next(const double* __restrict__ partial, const double* __restrict__ k2d,
                               const float* __restrict__ qaPrev, float* __restrict__ qaNext) {
  int k = threadIdx.x;
  double T = 0.0;
  for (int b = 0; b < NBLK; ++b) T += partial[(size_t)b * 128 + k];
  T = fmax(T, 1e-300);
  qaNext[k] = (float)(log2(k2d[k]) + (double)qaPrev[k] - log2(T));
}

// --- Middle/final passes.
// Per 16-row tile: q[b,k] = 20*log2e*L + qa[k]; p = row max; s = 2^(q-p) in [0,1].
// Row sums ctilde via WMMA (A = 16x4 s-fragment, B = ones -> every D column holds row sums).
// WRITE_OUT=false: accumulate T[k] += s/ctilde (f64, deterministic reduction).
// WRITE_OUT=true : out[b,k] = s/ctilde  (== E*k2/(R2*C3) exactly, scales cancel).
template <bool WRITE_OUT>
__global__ __launch_bounds__(THREADS) void k_pass(const float* __restrict__ logits,
                                                  const float* __restrict__ qa,
                                                  double* __restrict__ partial,
                                                  float* __restrict__ out) {
  __shared__ float tileL[8][16 * PITCH];  // per-wave 16x64 staging buffer (two K-halves reuse it)
  __shared__ double wred[8][128];
  __shared__ float qaS[128];
  if (threadIdx.x < 128) qaS[threadIdx.x] = qa[threadIdx.x];
  __syncthreads();

  const int w = threadIdx.x >> 5;
  const int l = threadIdx.x & 31;
  const int row = l & 15;          // WMMA A-layout: lane%16 <-> matrix row
  const int cofs = (l >> 4) * 2;   // low half lanes: K pair {0,1}; high half: {2,3}
  const int wgid = blockIdx.x * 8 + w;
  const float K1 = 28.8539008f;    // 20/ln2

  double colacc[64];
  if (!WRITE_OUT) {
#pragma unroll
    for (int m = 0; m < 64; ++m) colacc[m] = 0.0;
  }

  unsigned int ldsbase = (unsigned int)(unsigned long long)&tileL[w][0];

  for (int tile = wgid; tile < N_TILES; tile += N_WAVES) {
    const float* gp = logits + (size_t)tile * (16 * K_DIM);
    float qc[64];
    float mx = -3.0e38f;

#pragma unroll
    for (int h = 0; h < 2; ++h) {
      // async-stage 16 rows x 64 floats (4KB) of this K-half into LDS
#pragma unroll
      for (int it = 0; it < 8; ++it) {
        int r = it * 2 + (l >> 4);
        int c16 = l & 15;
        unsigned int lo = ldsbase + (unsigned int)(r * (PITCH * 4) + c16 * 16);
        const float* g = gp + r * K_DIM + h * 64 + c16 * 4;
        asm volatile("global_load_async_to_lds_b128 %0, %1, off" ::"v"(lo), "v"(g) : "memory");
      }
      asm volatile("s_wait_asynccnt 0x0" ::: "memory");
      // compute q for this half (A-layout strided LDS reads, conflict-free with PITCH=68)
#pragma unroll
      for (int j = 0; j < 16; ++j) {
        float2 Lp = *(const float2*)&tileL[w][row * PITCH + 4 * j + cofs];
        float2 qp = *(const float2*)&qaS[h * 64 + 4 * j + cofs];
        float q0 = fmaf(Lp.x, K1, qp.x);
        float q1 = fmaf(Lp.y, K1, qp.y);
        int m = h * 32 + 2 * j;
        qc[m] = q0;
        qc[m + 1] = q1;
        mx = fmaxf(mx, fmaxf(q0, q1));
      }
    }
    // row max lives split across lane l and l^16
    mx = fmaxf(mx, __shfl_xor(mx, 16, 32));

    v8f acc = {0.f, 0.f, 0.f, 0.f, 0.f, 0.f, 0.f, 0.f};
    v2f onesv;
    onesv[0] = 1.0f;
    onesv[1] = 1.0f;
#pragma unroll
    for (int m = 0; m < 32; ++m) {
      float s0 = fast_exp2(qc[2 * m] - mx);
      float s1 = fast_exp2(qc[2 * m + 1] - mx);
      qc[2 * m] = s0;
      qc[2 * m + 1] = s1;
      v2f a;
      a[0] = s0;
      a[1] = s1;
      // D = A(16x4) * ones(4x16) + C  -> every D column accumulates the 16 row sums
      acc = __builtin_amdgcn_wmma_f32_16x16x4_f32(false, a, false, onesv, (short)0, acc, false,
                                                  false);
    }

    // Extract ctilde for this lane's row from the D layout
    // (lane<16: VGPR v = rowsum[v]; lane>=16: VGPR v = rowsum[v+8])
    int r8 = l & 7;
    float tmp = acc[0];
    if (r8 == 1) tmp = acc[1];
    if (r8 == 2) tmp = acc[2];
    if (r8 == 3) tmp = acc[3];
    if (r8 == 4) tmp = acc[4];
    if (r8 == 5) tmp = acc[5];
    if (r8 == 6) tmp = acc[6];
    if (r8 == 7) tmp = acc[7];
    float swv = __shfl_xor(tmp, 16, 32);
    bool useSwap = (((l >> 3) & 1) != ((l >> 4) & 1));
    float ct = useSwap ? swv : tmp;  // ctilde >= 1 by construction
    float invc = 1.0f / ct;

    if (WRITE_OUT) {
      float* ob = out + (size_t)tile * (16 * K_DIM) + (size_t)row * K_DIM + cofs;
#pragma unroll
      for (int h = 0; h < 2; ++h) {
#pragma unroll
        for (int j = 0; j < 16; ++j) {
          int m = h * 32 + 2 * j;
          float2 o;
          o.x = qc[m] * invc;
          o.y = qc[m + 1] * invc;
          *(float2*)&ob[h * 64 + 4 * j] = o;
        }
      }
    } else {
#pragma unroll
      for (int m = 0; m < 64; ++m) colacc[m] += (double)(qc[m] * invc);
    }
  }

  if (!WRITE_OUT) {
    // deterministic reduction: 16-lane butterflies (lanes within a half hold the same k-set)
#pragma unroll
    for (int msk = 1; msk <= 8; msk <<= 1) {
#pragma unroll
      for (int m = 0; m < 64; ++m) colacc[m] += __shfl_xor(colacc[m], msk, 32);
    }
    if (row == 0) {  // lanes 0 and 16 write their half's 64 k values
#pragma unroll
      for (int m = 0; m < 64; ++m) {
        int h = m >> 5, jj = (m >> 1) & 15, b = m & 1;
        wred[w][h * 64 + 4 * jj + cofs + b] = colacc[m];
      }
    }
    __syncthreads();
    if (threadIdx.x < 128) {
      double s = 0.0;
#pragma unroll
      for (int i = 0; i < 8; ++i) s += wred[i][threadIdx.x];
      partial[(size_t)blockIdx.x * 128 + threadIdx.x] = s;
    }
  }
}

extern "C" void kernel_launch(void* const* d_in, const int* in_sizes, int n_in, void* d_out,
                              int out_size, void* d_ws, size_t ws_size, hipStream_t stream) {
  const float* logits = (const float*)d_in[0];  // [1048576, 128] f32
  const float* wvec = (const float*)d_in[1];    // [1, 128] f32
  float* out = (float*)d_out;                   // [1048576, 128]

  char* wsb = (char*)d_ws;
  double* k2d = (double*)wsb;               // 1024 B
  float* qa0 = (float*)(wsb + 1024);        // 512 B
  float* qa1 = (float*)(wsb + 1536);        // 512 B
  float* qa2 = (float*)(wsb + 2048);        // 512 B
  double* partial = (double*)(wsb + 4096);  // NBLK*128*8 = 1 MB (reused each pass)

  k_softmax_w<<<1, 32, 0, stream>>>(wvec, k2d);
  k_colsum0<<<NBLK, THREADS, 0, stream>>>(logits, partial);
  k_make_qa0<<<1, 128, 0, stream>>>(partial, k2d, qa0);

  k_pass<false><<<NBLK, THREADS, 0, stream>>>(logits, qa0, partial, nullptr);
  k_make_qa_next<<<1, 128, 0, stream>>>(partial, k2d, qa0, qa1);

  k_pass<false><<<NBLK, THREADS, 0, stream>>>(logits, qa1, partial, nullptr);
  k_make_qa_next<<<1, 128, 0, stream>>>(partial, k2d, qa1, qa2);

  k_pass<true><<<NBLK, THREADS, 0, stream>>>(logits, qa2, nullptr, out);
}